// SinPositionEmbedding_21388937134617
// MI455X (gfx1250) — compile-verified
//
#include <hip/hip_runtime.h>

// Sinusoidal position embedding: out[b, 2i]   = sin(t[b] * 10000^(-2i/D)),
//                                out[b, 2i+1] = cos(t[b] * 10000^(-2i/D))
// B = 16384, D = 2048 (reference setup). Pure store-bandwidth-bound:
// 134 MB out @ 23.3 TB/s ~= 5.8 us floor. No contraction dimension -> the
// optimal CDNA5 path is TRANS-pipe v_exp/v_sin/v_cos + global_store_b128,
// not WMMA.

namespace {

constexpr int   kD        = 2048;          // embedding dim (reference fixes this)
constexpr int   kRows     = 8;             // rows per block (amortizes exp2)
constexpr int   kTPB      = kD / 4;        // 512 threads: one float4 slot each per row
constexpr float kLog2N    = 13.287712379549449f;          // log2(10000)
constexpr float kFreqC    = -2.0f * kLog2N / (float)kD;   // inv_freq[p] = exp2(p * kFreqC)
constexpr float kInv2Pi   = 0.15915494309189535f;
constexpr float kTwoPiHi  = 6.2831854820251465f;          // float(2*pi)
constexpr float kTwoPiLo  = -1.7484556e-7f;               // 2*pi - kTwoPiHi

// Cody-Waite range reduction to |r| <= pi (radians), then convert to
// revolutions and use the native TRANS ops (v_sin_f32 / v_cos_f32 take
// input in revolutions). Worst-case error ~3e-7 rad even for angles up to
// t_max * 1.0 = 1000 rad.
__device__ __forceinline__ void sincos_fast(float ang, float& s, float& c) {
  const float k = rintf(ang * kInv2Pi);        // v_rndne_f32
  float r = fmaf(-k, kTwoPiHi, ang);           // single-rounded fma
  r = fmaf(-k, kTwoPiLo, r);
  const float rev = r * kInv2Pi;               // |rev| <= 0.5 revolutions
  s = __builtin_amdgcn_sinf(rev);              // v_sin_f32 (TRANS)
  c = __builtin_amdgcn_cosf(rev);              // v_cos_f32 (TRANS)
}

__device__ __forceinline__ void emit_row(float tb, float f0, float f1,
                                         float* __restrict__ rowp, int tid) {
  const float a0 = tb * f0;
  const float a1 = tb * f1;
  float4 v;
  sincos_fast(a0, v.x, v.y);                   // columns 4*tid, 4*tid+1
  sincos_fast(a1, v.z, v.w);                   // columns 4*tid+2, 4*tid+3
  // 16B-aligned coalesced store -> global_store_b128
  *reinterpret_cast<float4*>(rowp + 4 * tid) = v;
}

} // namespace

__global__ __launch_bounds__(kTPB)
void sinpos_embed_kernel(const float* __restrict__ t,
                         float* __restrict__ out,
                         int B) {
  const int tid = threadIdx.x;          // 0..511 : which float4 of the row
  const int p0  = 2 * tid;              // first frequency index of this slot

  // inv_freq for the two pairs this thread owns (computed once per 8 rows).
  const float f0 = __builtin_amdgcn_exp2f((float)p0 * kFreqC);        // v_exp_f32
  const float f1 = __builtin_amdgcn_exp2f((float)(p0 + 1) * kFreqC);  // v_exp_f32

  const int row0 = blockIdx.x * kRows;

  if (row0 + kRows <= B) {
    // Fast path (all blocks when B % kRows == 0): no per-row guards.
    // Pre-load the 8 block-uniform t values (scalar loads, mergeable),
    // then straight-line compute + 8 back-to-back global_store_b128.
    float tb[kRows];
#pragma unroll
    for (int r = 0; r < kRows; ++r) tb[r] = t[row0 + r];

#pragma unroll
    for (int r = 0; r < kRows; ++r) {
      emit_row(tb[r], f0, f1, out + (size_t)(row0 + r) * kD, tid);
    }
  } else {
    // Ragged tail (generic B only).
#pragma unroll
    for (int r = 0; r < kRows; ++r) {
      const int b = row0 + r;
      if (b < B) {
        emit_row(t[b], f0, f1, out + (size_t)b * kD, tid);
      }
    }
  }
}

extern "C" void kernel_launch(void* const* d_in, const int* in_sizes, int n_in,
                              void* d_out, int out_size, void* d_ws, size_t ws_size,
                              hipStream_t stream) {
  const float* t   = (const float*)d_in[0];  // [B] float32 timesteps
  float*       out = (float*)d_out;          // [B, 2048] float32
  const int B = in_sizes[0];                 // 16384 in the reference setup

  const int blocks = (B + kRows - 1) / kRows;
  sinpos_embed_kernel<<<blocks, kTPB, 0, stream>>>(t, out, B);
}